// BernoulliEdge_39814346834548
// MI455X (gfx1250) — compile-verified
//
#include <hip/hip_runtime.h>
#include <stdint.h>

typedef __attribute__((ext_vector_type(2))) float v2f;
typedef __attribute__((ext_vector_type(4))) float v4f;
typedef __attribute__((ext_vector_type(8))) float v8f;

#define N_NODES 1024
#define FDIM    128
#define KDIM    256   // 2F
#define BATCH   16

// ---------------------------------------------------------------------------
// threefry2x32 with JAX key(42) -> key data (hi, lo) = (0, 42)
// ---------------------------------------------------------------------------
__device__ __forceinline__ uint32_t rotl32(uint32_t v, int r) {
    return (v << r) | (v >> (32 - r));
}

__device__ __forceinline__ void threefry2x32_0_42(uint32_t& x0, uint32_t& x1) {
    const uint32_t ks0 = 0u, ks1 = 42u, ks2 = 0x1BD11BDAu ^ 0u ^ 42u;
    x0 += ks0; x1 += ks1;
#define TF_ROUND(r) { x0 += x1; x1 = rotl32(x1, (r)); x1 ^= x0; }
    TF_ROUND(13) TF_ROUND(15) TF_ROUND(26) TF_ROUND(6)
    x0 += ks1; x1 += ks2 + 1u;
    TF_ROUND(17) TF_ROUND(29) TF_ROUND(16) TF_ROUND(24)
    x0 += ks2; x1 += ks0 + 2u;
    TF_ROUND(13) TF_ROUND(15) TF_ROUND(26) TF_ROUND(6)
    x0 += ks0; x1 += ks1 + 3u;
    TF_ROUND(17) TF_ROUND(29) TF_ROUND(16) TF_ROUND(24)
    x0 += ks1; x1 += ks2 + 4u;
    TF_ROUND(13) TF_ROUND(15) TF_ROUND(26) TF_ROUND(6)
    x0 += ks2; x1 += ks0 + 5u;
#undef TF_ROUND
}

// JAX uniform: bitcast(bits>>9 | 0x3f800000) - 1 in [0,1), scaled to [1e-10, 1)
__device__ __forceinline__ float bits_to_uniform(uint32_t r) {
    float f = __uint_as_float((r >> 9) | 0x3f800000u) - 1.0f;
    f = f * (1.0f - 1e-10f) + 1e-10f;
    return fmaxf(f, 1e-10f);
}

// ---------------------------------------------------------------------------
// Kernel 1: probs[b,i] = clip(sigmoid(W2 . lrelu(W1^T [left|node_i] + b1) + b2))
// One wave per 16-row tile; V_WMMA_F32_16X16X4_F32 over K=256.
// ---------------------------------------------------------------------------
__global__ __launch_bounds__(128)
void edge_mlp_probs_kernel(const float* __restrict__ nodes,
                           const int*   __restrict__ num_nodes,
                           const float* __restrict__ W1,
                           const float* __restrict__ b1,
                           const float* __restrict__ W2,
                           const float* __restrict__ b2,
                           float*       __restrict__ probs) {
    __shared__ float sh[4][16][KDIM];   // 64 KB: per-wave 16x256 net_in tile

    const int lane = threadIdx.x & 31;
    const int wv   = threadIdx.x >> 5;
    const int tile = blockIdx.x * 4 + wv;        // 0..1023 (16384 rows / 16)
    const int b    = tile >> 6;                  // 64 tiles per batch
    const int i0   = (tile & 63) * 16;           // row base within batch

    const float* nodesB = nodes + (size_t)b * N_NODES * FDIM;
    const int    nn     = num_nodes[b];
    const float* left   = nodesB + (size_t)nn * FDIM;

    // Stage net_in tile [16 x 256] = [left(128) | nodes_row(128)] into LDS.
    for (int u = lane; u < 16 * 64; u += 32) {
        const int r = u >> 6;
        const int q = u & 63;
        const float4 v = (q < 32)
            ? *(const float4*)(left + (q << 2))
            : *(const float4*)(nodesB + (size_t)(i0 + r) * FDIM + ((q - 32) << 2));
        *(float4*)&sh[wv][r][q << 2] = v;
    }
    __syncthreads();

    const int half = lane >> 4;   // selects K pair {0,1} vs {2,3}
    const int ln   = lane & 15;   // A: row M;  B/C: column N

    const v8f zero8 = {0.f, 0.f, 0.f, 0.f, 0.f, 0.f, 0.f, 0.f};
    v8f acc[8];
#pragma unroll
    for (int t = 0; t < 8; ++t) acc[t] = zero8;

    for (int kb = 0; kb < KDIM; kb += 4) {
        const int k = kb + 2 * half;
        const float2 af = *(const float2*)&sh[wv][ln][k];   // ds_load_b64
        v2f A; A.x = af.x; A.y = af.y;                      // A[M=ln][K=k,k+1]
#pragma unroll
        for (int t = 0; t < 8; ++t) {
            v2f Bf;                                         // B[K][N=16t+ln]
            Bf.x = W1[(size_t)k       * FDIM + t * 16 + ln];
            Bf.y = W1[(size_t)(k + 1) * FDIM + t * 16 + ln];
            acc[t] = __builtin_amdgcn_wmma_f32_16x16x4_f32(
                false, A, false, Bf, (short)0, acc[t], false, false);
        }
    }

    // Epilogue: +b1, LeakyReLU(0.01), dot with W2, lane-half reduce, sigmoid.
    float b1v[8], w2v[8];
#pragma unroll
    for (int t = 0; t < 8; ++t) {
        b1v[t] = b1[t * 16 + ln];
        w2v[t] = W2[t * 16 + ln];
    }
    const float bias2 = b2[0];

    float part[8];   // rows M = 8*half + r
#pragma unroll
    for (int r = 0; r < 8; ++r) {
        float s = 0.0f;
#pragma unroll
        for (int t = 0; t < 8; ++t) {
            float h = acc[t][r] + b1v[t];
            h = (h > 0.0f) ? h : 0.01f * h;
            s += h * w2v[t];
        }
        part[r] = s;
    }
#pragma unroll
    for (int m = 1; m < 16; m <<= 1) {
#pragma unroll
        for (int r = 0; r < 8; ++r)
            part[r] += __shfl_xor(part[r], m, 32);
    }
    if (ln == 0) {
#pragma unroll
        for (int r = 0; r < 8; ++r) {
            const float z = part[r] + bias2;
            float p = 1.0f / (1.0f + __expf(-z));
            p = fminf(fmaxf(p, 0.001f), 0.999f);
            probs[tile * 16 + half * 8 + r] = p;
        }
    }
}

// ---------------------------------------------------------------------------
// Kernel 2: w = where(j==nn & i<nn, probs, weights); gumbel-hard sample.
// Streaming: NT loads of weights, NT b128 stores of both outputs.
// ---------------------------------------------------------------------------
__global__ __launch_bounds__(256)
void gumbel_sample_kernel(const float* __restrict__ probs,
                          const float* __restrict__ weights,
                          const int*   __restrict__ num_nodes,
                          float*       __restrict__ out) {
    constexpr uint32_t NN    = (uint32_t)N_NODES * N_NODES;  // 1M
    constexpr uint32_t TOTAL = (uint32_t)BATCH * NN;         // 16M

    const uint32_t e0  = (blockIdx.x * 256u + threadIdx.x) * 4u;
    const uint32_t b   = e0 >> 20;
    const uint32_t rem = e0 & (NN - 1u);
    const uint32_t i   = rem >> 10;
    const uint32_t j0  = rem & (N_NODES - 1u);

    const int   nn = num_nodes[b];                 // uniform per block (s_load)
    const float p  = probs[(b << 10) + i];
    const bool  row_ok = (int)i < nn;

    const v4f wv = __builtin_nontemporal_load((const v4f*)(weights + e0));

    float hard[4], wres[4];
#pragma unroll
    for (int q = 0; q < 4; ++q) {
        const uint32_t j = j0 + (uint32_t)q;
        const float w = (row_ok && (int)j == nn) ? p : wv[q];

        uint32_t x0 = e0 + (uint32_t)q;            // counter for class 0
        uint32_t x1 = x0 + TOTAL;                  // counter for class 1
        threefry2x32_0_42(x0, x1);
        const float u0 = bits_to_uniform(x0);
        const float u1 = bits_to_uniform(x1);
        const float g0 = -__logf(-__logf(u0));
        const float g1 = -__logf(-__logf(u1));

        const float l1 = __logf(fminf(fmaxf(w,        0.001f), 0.999f));
        const float l0 = __logf(fminf(fmaxf(1.0f - w, 0.001f), 0.999f));
        // argmax ties resolve to class 0 -> strict compare for hard==1
        hard[q] = (l1 + g1 > l0 + g0) ? 1.0f : 0.0f;
        wres[q] = w;
    }

    v4f h4 = {hard[0], hard[1], hard[2], hard[3]};
    v4f w4 = {wres[0], wres[1], wres[2], wres[3]};
    __builtin_nontemporal_store(h4, (v4f*)(out + e0));          // adj_out
    __builtin_nontemporal_store(w4, (v4f*)(out + TOTAL + e0));  // w
}

// ---------------------------------------------------------------------------
extern "C" void kernel_launch(void* const* d_in, const int* in_sizes, int n_in,
                              void* d_out, int out_size, void* d_ws, size_t ws_size,
                              hipStream_t stream) {
    const float* nodes     = (const float*)d_in[0];
    // d_in[1] = adj (overwritten by module; unused)
    const float* weights   = (const float*)d_in[2];
    const int*   num_nodes = (const int*)  d_in[3];
    // d_in[4] = B scalar (16, baked into constants)
    const float* W1 = (const float*)d_in[5];
    const float* b1 = (const float*)d_in[6];
    const float* W2 = (const float*)d_in[7];
    const float* b2 = (const float*)d_in[8];

    float* out   = (float*)d_out;
    float* probs = (float*)d_ws;   // BATCH*N_NODES floats = 64 KB scratch

    // 1024 row-tiles of 16 rows, 4 waves/block -> 256 blocks
    edge_mlp_probs_kernel<<<256, 128, 0, stream>>>(
        nodes, num_nodes, W1, b1, W2, b2, probs);

    // 16M elements / (256 threads * 4 elems) = 16384 blocks
    constexpr uint32_t TOTAL = (uint32_t)BATCH * N_NODES * N_NODES;
    gumbel_sample_kernel<<<TOTAL / (256u * 4u), 256, 0, stream>>>(
        probs, weights, num_nodes, out);
}